// APPNP_73710228734491
// MI455X (gfx1250) — compile-verified
//
#include <hip/hip_runtime.h>

// ---------------- APPNP propagation on MI455X (gfx1250) ----------------
// K=5, ALPHA=0.8, N=100000 nodes, E=1.6M edges, D=64 feats (fp32).
// L2-resident scatter/gather SpMM: bound by L2 atomics + gathers, not FLOPs.
// CDNA5 path used: async global->LDS staging of edge tiles (ASYNCcnt) +
// native fp32 global atomics. WMMA does not apply (no dense contraction).

#define ALPHA_C   0.8f
#define KITER     5
#define DFEAT     64
#define BLK       256          // 8 wave32 waves per block
#define TILE      1024         // edges staged per block (= BLK * 4)
#define WAVES_PB  (BLK / 32)

#if defined(__HIP_DEVICE_COMPILE__) && \
    __has_builtin(__builtin_amdgcn_global_load_async_to_lds_b128) && \
    __has_builtin(__builtin_amdgcn_s_wait_asynccnt)
#define USE_ASYNC 1
#else
#define USE_ASYNC 0
#endif

#if USE_ASYNC
typedef int v4i __attribute__((vector_size(16)));
typedef __attribute__((address_space(1))) v4i gv4i_t;   // global (AS1)
typedef __attribute__((address_space(3))) v4i lv4i_t;   // LDS (AS3)

__device__ __forceinline__ void async_b128(const void* g, void* l) {
  // global_load_async_to_lds_b128: (global src, LDS dst, imm offset, imm cpol)
  __builtin_amdgcn_global_load_async_to_lds_b128((gv4i_t*)g, (lv4i_t*)l, 0, 0);
}
#endif

__global__ void __launch_bounds__(BLK) k_deg_init(float* __restrict__ deg, int n) {
  int i = blockIdx.x * BLK + threadIdx.x;
  if (i < n) deg[i] = 1.0f;                 // self-loop contribution
}

__global__ void __launch_bounds__(BLK) k_deg_accum(const int* __restrict__ dst,
                                                   float* __restrict__ deg, int e) {
  int i = blockIdx.x * BLK + threadIdx.x;
  if (i < e) unsafeAtomicAdd(&deg[dst[i]], 1.0f);
}

__global__ void __launch_bounds__(BLK) k_rsqrt_inplace(float* __restrict__ d, int n) {
  int i = blockIdx.x * BLK + threadIdx.x;
  if (i < n) d[i] = rsqrtf(d[i]);           // deg >= 1 always (self-loops)
}

__global__ void __launch_bounds__(BLK) k_edge_norm(const int* __restrict__ src,
                                                   const int* __restrict__ dst,
                                                   const float* __restrict__ dis,
                                                   float* __restrict__ nrm, int e) {
  int i = blockIdx.x * BLK + threadIdx.x;
  if (i < e) nrm[i] = (1.0f - ALPHA_C) * dis[src[i]] * dis[dst[i]];  // fold 0.2
}

// y = ALPHA*h0 + (1-ALPHA)*(1/deg)*x   (teleport + self-loop term, float4)
__global__ void __launch_bounds__(BLK) k_seed(const float* __restrict__ xin,
                                              const float* __restrict__ h0,
                                              const float* __restrict__ dis,
                                              float* __restrict__ y, int n4) {
  int i = blockIdx.x * BLK + threadIdx.x;
  if (i >= n4) return;
  int node = i >> 4;                        // 16 float4 per node row
  float di = dis[node];
  float sw = (1.0f - ALPHA_C) * di * di;    // = 0.2 / deg
  float4 xv = ((const float4*)xin)[i];
  float4 hv = ((const float4*)h0)[i];
  float4 r;
  r.x = fmaf(sw, xv.x, ALPHA_C * hv.x);
  r.y = fmaf(sw, xv.y, ALPHA_C * hv.y);
  r.z = fmaf(sw, xv.z, ALPHA_C * hv.z);
  r.w = fmaf(sw, xv.w, ALPHA_C * hv.w);
  ((float4*)y)[i] = r;
}

// y[dst] += norm_e * x[src]  over a 1024-edge tile per block.
// Metadata staged to LDS (async if available). Each half-wave owns one edge:
// 16 lanes x float4 = full 64-float row -> 1 b128 gather + 4 f32 atomics/lane.
__global__ void __launch_bounds__(BLK) k_scatter(const int* __restrict__ src,
                                                 const int* __restrict__ dst,
                                                 const float* __restrict__ nrm,
                                                 const float* __restrict__ x,
                                                 float* __restrict__ y, int e) {
  __shared__ int   ls[TILE];
  __shared__ int   ld[TILE];
  __shared__ float lw[TILE];

  const int base = blockIdx.x * TILE;
  const int cnt  = min(TILE, e - base);
  const int t    = threadIdx.x;
  const int idx  = t * 4;                   // TILE == BLK*4 -> one pass

#if USE_ASYNC
  if (base + idx + 4 <= e) {
    async_b128(src + base + idx, &ls[idx]);
    async_b128(dst + base + idx, &ld[idx]);
    async_b128(nrm + base + idx, &lw[idx]);
  } else {
    for (int j = 0; j < 4; ++j) {
      int ii = idx + j;
      if (ii < cnt) { ls[ii] = src[base+ii]; ld[ii] = dst[base+ii]; lw[ii] = nrm[base+ii]; }
    }
  }
  __builtin_amdgcn_s_wait_asynccnt(0);
  __syncthreads();
#else
  for (int j = 0; j < 4; ++j) {
    int ii = idx + j;
    if (ii < cnt) { ls[ii] = src[base+ii]; ld[ii] = dst[base+ii]; lw[ii] = nrm[base+ii]; }
  }
  __syncthreads();
#endif

  const int wave = t >> 5;
  const int lane = t & 31;
  const int sub  = lane >> 4;               // half-wave id: 0 or 1
  const int fo   = (lane & 15) * 4;         // feature offset (float4 granule)
  const int per  = TILE / WAVES_PB;         // 128 edges per wave
  const int begin = wave * per;
  const int end   = min(begin + per, cnt);

#pragma unroll 4
  for (int i = begin; i < end; i += 2) {
    int eidx = i + sub;
    if (eidx < end) {
      int   s = ls[eidx];
      int   d = ld[eidx];
      float w = lw[eidx];
      const float4 v = *(const float4*)(x + (size_t)s * DFEAT + fo);
      float* yr = y + (size_t)d * DFEAT + fo;
      unsafeAtomicAdd(yr + 0, w * v.x);
      unsafeAtomicAdd(yr + 1, w * v.y);
      unsafeAtomicAdd(yr + 2, w * v.z);
      unsafeAtomicAdd(yr + 3, w * v.w);
    }
  }
}

static inline size_t align256(size_t x) { return (x + 255) & ~(size_t)255; }

extern "C" void kernel_launch(void* const* d_in, const int* in_sizes, int n_in,
                              void* d_out, int out_size, void* d_ws, size_t ws_size,
                              hipStream_t stream) {
  (void)n_in; (void)out_size; (void)ws_size;
  const float* x  = (const float*)d_in[0];     // [N, 64] fp32, also h0
  const int*   ei = (const int*)d_in[1];       // [2, E] int32
  const int N = in_sizes[0] / DFEAT;
  const int E = in_sizes[1] / 2;
  const int* srcp = ei;                        // row 0: message sources
  const int* dstp = ei + E;                    // row 1: aggregation targets

  // Workspace: dis[N] | norm[E] | buf[N*64]  (~32.4 MB total)
  char* wsb = (char*)d_ws;
  size_t off = 0;
  float* dis = (float*)(wsb + off); off += align256((size_t)N * sizeof(float));
  float* nrm = (float*)(wsb + off); off += align256((size_t)E * sizeof(float));
  float* buf = (float*)(wsb + off);
  float* out = (float*)d_out;

  const dim3 b(BLK);
  k_deg_init     <<<dim3((N + BLK - 1) / BLK), b, 0, stream>>>(dis, N);
  k_deg_accum    <<<dim3((E + BLK - 1) / BLK), b, 0, stream>>>(dstp, dis, E);
  k_rsqrt_inplace<<<dim3((N + BLK - 1) / BLK), b, 0, stream>>>(dis, N);
  k_edge_norm    <<<dim3((E + BLK - 1) / BLK), b, 0, stream>>>(srcp, dstp, dis, nrm, E);

  // Ping-pong so iteration K lands in d_out: x -> out -> buf -> out -> buf -> out
  const int n4 = N * (DFEAT / 4);
  const float* cur = x;
  for (int k = 0; k < KITER; ++k) {
    float* nxt = (k % 2 == 0) ? out : buf;
    k_seed   <<<dim3((n4 + BLK - 1) / BLK), b, 0, stream>>>(cur, x, dis, nxt, n4);
    k_scatter<<<dim3((E + TILE - 1) / TILE), b, 0, stream>>>(srcp, dstp, nrm, cur, nxt, E);
    cur = nxt;
  }
}